// RoiPooling_40235253629499
// MI455X (gfx1250) — compile-verified
//
#include <hip/hip_runtime.h>
#include <cstdint>

typedef float floatx4 __attribute__((ext_vector_type(4)));

#define POOLK 7
#define FM_H  128
#define FM_W  128
#define FM_C  512

// One block per (roi, pooled-row). 128 threads: thread t owns channels [4t, 4t+4).
// Pipeline over the 7 pooled columns: async-stage the 4 corner lines (tl,tr,bl,br)
// for px+1 into LDS buffer (px+1)&1 while blending px from buffer px&1.
__global__ __launch_bounds__(128) void roi_pool_async_kernel(
    const float* __restrict__ fm,
    const float* __restrict__ rois,
    float* __restrict__ out)
{
    __shared__ float lds[2 * 4 * FM_C];   // 2 buffers x 4 lines x 512 f32 = 16 KB

    const int b   = blockIdx.x;
    const int roi = b / POOLK;
    const int py  = b - roi * POOLK;
    const int t   = threadIdx.x;

    // ROI box: rois[r] = [xmin, xmax, ymin, ymax] as integral floats
    const float* rp  = rois + (size_t)roi * 4;
    const int   xmin = (int)rp[0];
    const int   xmax = (int)rp[1];
    const int   ymin = (int)rp[2];
    const int   ymax = (int)rp[3];
    const int   sx   = xmax - xmin;
    const int   sy   = ymax - ymin;

    // y coordinates for this pooled row (matches _axis_coords in fp32)
    const float scy  = (float)sy / 7.0f;
    const float srcy = (float)py * scy;
    const int   iy0  = (int)floorf(srcy);
    const int   iy1  = min(iy0 + 1, sy - 1);
    const float fy   = srcy - (float)iy0;
    const int   y0   = ymin + iy0;
    const int   y1   = ymin + iy1;

    // x coordinates for all 7 pooled columns
    int   xs0[POOLK], xs1[POOLK];
    float fxs[POOLK];
    const float scx = (float)sx / 7.0f;
#pragma unroll
    for (int px = 0; px < POOLK; ++px) {
        const float s  = (float)px * scx;
        const int   i0 = (int)floorf(s);
        const int   i1 = min(i0 + 1, sx - 1);
        xs0[px] = xmin + i0;
        xs1[px] = xmin + i1;
        fxs[px] = s - (float)i0;
    }

    const int      c4      = t * 4;   // base channel for this thread
    const uint32_t ldsBase = (uint32_t)(uintptr_t)(&lds[0]); // low 32 bits = LDS byte addr

    // Issue 4 async b128 line loads (tl,tr,bl,br) for pooled column px into buffer buf.
    // Per-wave ASYNCcnt += 4.
    auto stage = [&](int px, int buf) {
#pragma unroll
        for (int r = 0; r < 4; ++r) {
            const int y = (r < 2) ? y0 : y1;
            const int x = (r & 1) ? xs1[px] : xs0[px];
            const float*   g = fm + ((size_t)(y * FM_W + x) * FM_C + c4);
            const uint32_t l = ldsBase +
                (uint32_t)(((buf * 4 + r) * FM_C + c4) * sizeof(float));
            asm volatile("global_load_async_to_lds_b128 %0, %1, off"
                         :: "v"(l), "v"(g)
                         : "memory");
        }
    };

    stage(0, 0);

#pragma unroll
    for (int px = 0; px < POOLK; ++px) {
        const int buf = px & 1;
        if (px < POOLK - 1) {
            stage(px + 1, buf ^ 1);
            // In-order async completion: <=4 outstanding => buffer `buf` is ready.
            asm volatile("s_wait_asynccnt 4" ::: "memory");
        } else {
            asm volatile("s_wait_asynccnt 0" ::: "memory");
        }

        const floatx4* L    = (const floatx4*)lds;
        const int      base = (buf * 4) * (FM_C / 4) + t;
        const floatx4  tl   = L[base];
        const floatx4  tr   = L[base + 1 * (FM_C / 4)];
        const floatx4  bl   = L[base + 2 * (FM_C / 4)];
        const floatx4  br   = L[base + 3 * (FM_C / 4)];

        const float fx = fxs[px];
        const floatx4 top = tl + (tr - tl) * fx;
        const floatx4 bot = bl + (br - bl) * fx;
        const floatx4 res = top + (bot - top) * fy;

        floatx4* op = (floatx4*)(out +
            (((size_t)(roi * POOLK + py) * POOLK + px) * FM_C + c4));
        __builtin_nontemporal_store(res, op);   // write-once stream: keep L2 for the feature map
    }
}

extern "C" void kernel_launch(void* const* d_in, const int* in_sizes, int n_in,
                              void* d_out, int out_size, void* d_ws, size_t ws_size,
                              hipStream_t stream) {
    const float* fm   = (const float*)d_in[0];   // (1,128,128,512) f32
    const float* rois = (const float*)d_in[1];   // (R,4) f32
    float*       out  = (float*)d_out;           // (R,7,7,512) f32

    const int R = in_sizes[1] / 4;
    roi_pool_async_kernel<<<dim3(R * POOLK), dim3(128), 0, stream>>>(fm, rois, out);
}